// BalanceLabels_77180562309196
// MI455X (gfx1250) — compile-verified
//
#include <hip/hip_runtime.h>
#include <hip/hip_bf16.h>
#include <stdint.h>

typedef float v4f __attribute__((ext_vector_type(4)));
typedef int   v4i __attribute__((ext_vector_type(4)));

// ---- problem geometry (fixed by the reference) ----
#define TOTAL_ELEMS     33554432u   // 4*128*256*256
#define SLAB_ELEMS      4194304u    // 1*64*256*256 -> 8 contiguous slabs
#define NSLABS          8
#define NCLASS          5
#define THREADS         256
#define BLOCKS_PER_SLAB 256
#define P_BLOCKS        2048        // NSLABS * BLOCKS_PER_SLAB
#define BLOCK_ELEMS     16384       // SLAB_ELEMS / BLOCKS_PER_SLAB
#define TILE            2048        // elements per async tile
#define NTILES          8           // BLOCK_ELEMS / TILE
#define NBUF            3           // triple buffer: prefetch depth 2
#define PARTIAL_U32S    (P_BLOCKS * 6)  // 12288 u32 = 48 KB (lives in d_out tail)

// Per-lane async copy: copies 16 bytes from the global address gaddr to the LDS address lds_addr, ASYNCcnt-tracked.
__device__ __forceinline__ void async_copy_b128(uint32_t lds_addr, uint64_t gaddr) {
  asm volatile("global_load_async_to_lds_b128 %0, %1, off"
               :: "v"(lds_addr), "v"(gaddr) : "memory");
}

// Histogram packed as five 12-bit fields in a u64; masksum in f32.
__device__ __forceinline__ void accum_one(int lab, float m,
                                          unsigned long long& acc, float& msum) {
  msum += m;
  unsigned one = (m > 0.0f) ? 1u : 0u;
  acc += ((unsigned long long)one) << (lab * 12);
}

// ---------------- pass 1: per-block masked histogram + mask sum ----------------
__global__ __launch_bounds__(THREADS)
void k_slab_reduce(const int* __restrict__ labels,
                   const float* __restrict__ mask,
                   unsigned* __restrict__ partials)
{
  alignas(16) __shared__ int   s_lab[NBUF][TILE];   // 24 KB
  alignas(16) __shared__ float s_msk[NBUF][TILE];   // 24 KB
  __shared__ unsigned s_wc[THREADS / 32][NCLASS];
  __shared__ float    s_wm[THREADS / 32];

  const int b    = blockIdx.x;
  const int t    = threadIdx.x;
  const int wave = t >> 5;
  const int lane = t & 31;

  const size_t base = (size_t)b * BLOCK_ELEMS;
  const uint64_t gl = (uint64_t)(uintptr_t)(labels + base);
  const uint64_t gm = (uint64_t)(uintptr_t)(mask   + base);
  // flat->LDS: low 32 bits of the flat address are the LDS byte offset
  const uint32_t llab = (uint32_t)(uintptr_t)&s_lab[0][0];
  const uint32_t lmsk = (uint32_t)(uintptr_t)&s_msk[0][0];
  // wave-private share of a tile: 256 elems = 1024 B; 2x b128 per array per wave
  const uint32_t shareOff = (uint32_t)(wave * 1024 + lane * 16);

  auto issue_tile = [&](int tt, int buf) {
    const uint64_t gOff = (uint64_t)tt * (TILE * 4);
    const uint32_t bOff = (uint32_t)buf * (TILE * 4);
#pragma unroll
    for (int j = 0; j < 2; ++j) {
      const uint32_t o = shareOff + (uint32_t)(j * 512);
      async_copy_b128(llab + bOff + o, gl + gOff + o);  // 4 async ops / wave / tile
      async_copy_b128(lmsk + bOff + o, gm + gOff + o);
    }
  };

  issue_tile(0, 0);
  issue_tile(1, 1);

  unsigned long long acc = 0ull;
  float msum = 0.0f;

#pragma unroll
  for (int tt = 0; tt < NTILES; ++tt) {
    const int buf = tt % NBUF;
    if (tt + 2 < NTILES) {
      issue_tile(tt + 2, (tt + 2) % NBUF);
      // tiles tt+1, tt+2 pending (8 ops) -> tile tt guaranteed complete
      asm volatile("s_wait_asynccnt 8" ::: "memory");
    } else if (tt + 1 < NTILES) {
      asm volatile("s_wait_asynccnt 4" ::: "memory");
    } else {
      asm volatile("s_wait_asynccnt 0" ::: "memory");
    }
    const int e = wave * 256 + lane * 8;   // each lane consumes 8 elements
    const v4i* lp4 = (const v4i*)&s_lab[buf][e];
    const v4f* mp4 = (const v4f*)&s_msk[buf][e];
    v4i L0 = lp4[0], L1 = lp4[1];
    v4f M0 = mp4[0], M1 = mp4[1];
    accum_one(L0.x, M0.x, acc, msum);
    accum_one(L0.y, M0.y, acc, msum);
    accum_one(L0.z, M0.z, acc, msum);
    accum_one(L0.w, M0.w, acc, msum);
    accum_one(L1.x, M1.x, acc, msum);
    accum_one(L1.y, M1.y, acc, msum);
    accum_one(L1.z, M1.z, acc, msum);
    accum_one(L1.w, M1.w, acc, msum);
  }

  // wave32 tree reduction, fixed order -> deterministic.
  // Per-lane field max = 64, per-wave max = 2048 < 4096 -> no 12-bit overflow.
#pragma unroll
  for (int off = 16; off > 0; off >>= 1) {
    unsigned lo  = (unsigned)acc;
    unsigned hi  = (unsigned)(acc >> 32);
    unsigned plo = __shfl_down(lo, off, 32);
    unsigned phi = __shfl_down(hi, off, 32);
    acc  += (((unsigned long long)phi) << 32) | (unsigned long long)plo;
    msum += __shfl_down(msum, off, 32);
  }
  if (lane == 0) {
#pragma unroll
    for (int c = 0; c < NCLASS; ++c)
      s_wc[wave][c] = (unsigned)((acc >> (12 * c)) & 0xFFFull);
    s_wm[wave] = msum;
  }
  __syncthreads();
  if (t == 0) {
    unsigned C[NCLASS] = {0, 0, 0, 0, 0};
    float ms = 0.0f;
#pragma unroll
    for (int w = 0; w < THREADS / 32; ++w) {
#pragma unroll
      for (int c = 0; c < NCLASS; ++c) C[c] += s_wc[w][c];
      ms += s_wm[w];
    }
    unsigned* p = partials + (size_t)b * 6;
#pragma unroll
    for (int c = 0; c < NCLASS; ++c) p[c] = C[c];
    p[5] = __float_as_uint(ms);
  }
}

// ---------------- pass 2: fold partials -> per-slab weights ----------------
__global__ __launch_bounds__(THREADS)
void k_weights(const unsigned* __restrict__ partials, float* __restrict__ weights)
{
  const int t = threadIdx.x;
  const int s = t >> 5;   // slab 0..7
  const int j = t & 31;   // lane within slab group
  unsigned c0 = 0, c1 = 0, c2 = 0, c3 = 0, c4 = 0;
  double ms = 0.0;
  for (int k = j; k < BLOCKS_PER_SLAB; k += 32) {
    const unsigned* p = partials + (size_t)(s * BLOCKS_PER_SLAB + k) * 6;
    c0 += p[0]; c1 += p[1]; c2 += p[2]; c3 += p[3]; c4 += p[4];
    ms += (double)__uint_as_float(p[5]);
  }
#pragma unroll
  for (int off = 16; off > 0; off >>= 1) {
    c0 += __shfl_down(c0, off, 32);
    c1 += __shfl_down(c1, off, 32);
    c2 += __shfl_down(c2, off, 32);
    c3 += __shfl_down(c3, off, 32);
    c4 += __shfl_down(c4, off, 32);
    ms += __shfl_down(ms, off, 32);
  }
  if (j == 0) {
    const unsigned cc[NCLASS] = {c0, c1, c2, c3, c4};
    const double denom = (ms > 0.0) ? ms : 1.0;
#pragma unroll
    for (int c = 0; c < NCLASS; ++c) {
      double f = (ms > 0.0) ? ((double)cc[c] / denom) : 0.0;
      f = (f < 0.05) ? 0.05 : ((f > 0.95) ? 0.95 : f);
      weights[s * NCLASS + c] = (float)(0.2 / f);   // (1/NUM_CLASSES)/frac
    }
  }
}

// ---------------- pass 3: out = mask * w[slab][label] ----------------
__global__ __launch_bounds__(THREADS)
void k_apply(const int* __restrict__ labels, const float* __restrict__ mask,
             const float* __restrict__ weights, float* __restrict__ out)
{
  __shared__ float s_w[NCLASS];   // banks 0..4 -> conflict-free multicast
  const int b = (P_BLOCKS - 1) - blockIdx.x;  // reverse: hit L2-resident tail first
  const int slab = b / BLOCKS_PER_SLAB;
  if (threadIdx.x < NCLASS) s_w[threadIdx.x] = weights[slab * NCLASS + threadIdx.x];
  __syncthreads();
  const size_t base = (size_t)b * BLOCK_ELEMS;
  const v4i* lp = (const v4i*)(labels + base);
  const v4f* mp = (const v4f*)(mask + base);
  v4f*       op = (v4f*)(out + base);
#pragma unroll 8
  for (int i = threadIdx.x; i < BLOCK_ELEMS / 4; i += THREADS) {
    v4i L = lp[i];
    v4f M = mp[i];
    v4f R;
    R.x = M.x * s_w[L.x];
    R.y = M.y * s_w[L.y];
    R.z = M.z * s_w[L.z];
    R.w = M.w * s_w[L.w];
    __builtin_nontemporal_store(R, op + i);   // write-once output, NT hint
  }
}

extern "C" void kernel_launch(void* const* d_in, const int* in_sizes, int n_in,
                              void* d_out, int out_size, void* d_ws, size_t ws_size,
                              hipStream_t stream) {
  (void)in_sizes; (void)n_in; (void)out_size; (void)ws_size;
  const int*   labels = (const int*)d_in[0];
  const float* mask   = (const float*)d_in[1];
  float* out = (float*)d_out;
  // Block partials live in the tail of d_out; k_apply overwrites them with the
  // real output afterwards. Only the 160-byte weight table needs d_ws.
  unsigned* partials = (unsigned*)(out + (TOTAL_ELEMS - PARTIAL_U32S));
  float*    weights  = (float*)d_ws;

  k_slab_reduce<<<P_BLOCKS, THREADS, 0, stream>>>(labels, mask, partials);
  k_weights<<<1, THREADS, 0, stream>>>(partials, weights);
  k_apply<<<P_BLOCKS, THREADS, 0, stream>>>(labels, mask, weights, out);
}